// MultiHeadGraphAttention_25357486915839
// MI455X (gfx1250) — compile-verified
//
#include <hip/hip_runtime.h>
#include <hip/hip_bf16.h>

typedef float v2f __attribute__((ext_vector_type(2)));
typedef float v8f __attribute__((ext_vector_type(8)));

#define F_OUT 128
#define N_HEAD 4

// ---------------------------------------------------------------------------
// Build B (128 x 16, row-major, K-major): col j<4 = w[j]*a_src[j],
// col 4..7 = w[j-4]*a_dst[j-4], cols 8..15 = 0 (pad to WMMA N=16).
// w is (H,1,F) flat; a is (H,2F,1) flat.
// ---------------------------------------------------------------------------
__global__ void build_b_kernel(const float* __restrict__ w,
                               const float* __restrict__ a,
                               float* __restrict__ Bmat) {
  for (int i = threadIdx.x; i < F_OUT * 16; i += blockDim.x) {
    int k = i >> 4;
    int n = i & 15;
    float v = 0.f;
    if (n < N_HEAD) {
      v = w[n * F_OUT + k] * a[n * 2 * F_OUT + k];              // a_src
    } else if (n < 2 * N_HEAD) {
      int h = n - N_HEAD;
      v = w[h * F_OUT + k] * a[h * 2 * F_OUT + F_OUT + k];      // a_dst
    }
    Bmat[i] = v;
  }
}

// ---------------------------------------------------------------------------
// Score GEMM via V_WMMA_F32_16X16X4_F32: one wave32 per 16-node tile.
//   A (16x4 f32): lanes 0-15 -> M, VGPR0/1 = K0/K1; lanes 16-31 -> K2/K3.
//   B (4x16 f32): v0 = K{0+hi}, v1 = K{2+hi}, N = lane&15 (from LDS).
//   C/D (16x16 f32): VGPR j -> M = j + 8*hi, N = lane&15.
// D columns 0..3 -> s_src[head], 4..7 -> s_dst[head].
// ---------------------------------------------------------------------------
__global__ void score_wmma_kernel(const float* __restrict__ x,
                                  const float* __restrict__ Bmat,
                                  float* __restrict__ s_src,
                                  float* __restrict__ s_dst,
                                  int N) {
  __shared__ float Bs[F_OUT * 16];
  for (int i = threadIdx.x; i < F_OUT * 16; i += blockDim.x) Bs[i] = Bmat[i];
  __syncthreads();

  const int lane = threadIdx.x & 31;
  const int wave = threadIdx.x >> 5;
  const int numTiles = (N + 15) >> 4;
  const int tile = blockIdx.x * (blockDim.x >> 5) + wave;
  if (tile >= numTiles) return;
  const int base = tile << 4;

  const int mrow = lane & 15;   // M index (A) and N index (B/C/D)
  const int hi   = lane >> 4;   // half-wave selector
  int arowi = base + mrow;
  if (arowi >= N) arowi = N - 1;                 // clamp tail reads
  const size_t arow = (size_t)arowi * F_OUT + 2 * hi;

  v8f c = {0.f, 0.f, 0.f, 0.f, 0.f, 0.f, 0.f, 0.f};
#pragma unroll
  for (int kk = 0; kk < F_OUT / 4; ++kk) {
    v2f av = *(const v2f*)(x + arow + 4 * kk);
    v2f bv;
    bv.x = Bs[(4 * kk + hi) * 16 + mrow];
    bv.y = Bs[(4 * kk + 2 + hi) * 16 + mrow];
    c = __builtin_amdgcn_wmma_f32_16x16x4_f32(
        /*neg_a=*/false, av, /*neg_b=*/false, bv,
        /*c_mod=*/(short)0, c, /*reuse_a=*/false, /*reuse_b=*/false);
  }

  if (mrow < 2 * N_HEAD) {
#pragma unroll
    for (int j = 0; j < 8; ++j) {
      int node = base + j + 8 * hi;
      if (node < N) {
        float v = c[j];
        if (mrow < N_HEAD)
          s_src[(size_t)mrow * N + node] = v;
        else
          s_dst[(size_t)(mrow - N_HEAD) * N + node] = v;
      }
    }
  }
}

// ---------------------------------------------------------------------------
// Edge scatter: one wave32 per edge (grid-stride). Lanes 0-3 own one head
// each: score -> leaky_relu -> exp(-.) -> rowsum atomic. Broadcast the 4
// edge weights, then every lane handles 4 features with f32 HW atomics.
// h[head][dst][f] = x[dst][f]*w[head][f] computed on the fly (never stored).
// ---------------------------------------------------------------------------
__global__ void edge_kernel(const float* __restrict__ x,
                            const float* __restrict__ w,
                            const float* __restrict__ s_src,
                            const float* __restrict__ s_dst,
                            const int* __restrict__ src,
                            const int* __restrict__ dst,
                            float* __restrict__ rowsum,
                            float* __restrict__ out,
                            int E, int N) {
  const int lane = threadIdx.x & 31;
  const long long wid =
      ((long long)blockIdx.x * blockDim.x + threadIdx.x) >> 5;
  const long long nwaves = ((long long)gridDim.x * blockDim.x) >> 5;

  float4 wv[N_HEAD];
#pragma unroll
  for (int h = 0; h < N_HEAD; ++h)
    wv[h] = *(const float4*)(w + h * F_OUT + 4 * lane);

  for (long long e = wid; e < E; e += nwaves) {
    const int s = src[e];
    const int d = dst[e];

    float ee = 0.f;
    if (lane < N_HEAD) {
      float sc = s_src[(size_t)lane * N + s] + s_dst[(size_t)lane * N + d];
      float lr = sc > 0.f ? sc : 0.2f * sc;   // leaky_relu slope 0.2
      ee = expf(-lr);
      unsafeAtomicAdd(rowsum + (size_t)lane * N + s, ee);
    }
    float eh[N_HEAD];
#pragma unroll
    for (int h = 0; h < N_HEAD; ++h) eh[h] = __shfl(ee, h, 32);

    float4 xv = *(const float4*)(x + (size_t)d * F_OUT + 4 * lane);
#pragma unroll
    for (int h = 0; h < N_HEAD; ++h) {
      float cc = eh[h];
      float* o = out + ((size_t)h * N + s) * F_OUT + 4 * lane;
      unsafeAtomicAdd(o + 0, cc * wv[h].x * xv.x);
      unsafeAtomicAdd(o + 1, cc * wv[h].y * xv.y);
      unsafeAtomicAdd(o + 2, cc * wv[h].z * xv.z);
      unsafeAtomicAdd(o + 3, cc * wv[h].w * xv.w);
    }
  }
}

// ---------------------------------------------------------------------------
// out[h,n,:] /= rowsum[h,n]   (0/0 -> NaN, matching the reference)
// ---------------------------------------------------------------------------
__global__ void norm_kernel(float* __restrict__ out,
                            const float* __restrict__ rowsum, int N) {
  size_t idx = (size_t)blockIdx.x * blockDim.x + threadIdx.x;
  size_t total = (size_t)N_HEAD * N * F_OUT;
  if (idx >= total) return;
  size_t per_head = (size_t)N * F_OUT;
  size_t h = idx / per_head;
  size_t n = (idx % per_head) / F_OUT;
  out[idx] = out[idx] / rowsum[h * N + n];
}

extern "C" void kernel_launch(void* const* d_in, const int* in_sizes, int n_in,
                              void* d_out, int out_size, void* d_ws,
                              size_t ws_size, hipStream_t stream) {
  const float* x = (const float*)d_in[0];   // (N, 128) f32
  const float* w = (const float*)d_in[1];   // (4, 1, 128) f32
  const float* a = (const float*)d_in[2];   // (4, 256, 1) f32
  const int* ei  = (const int*)d_in[3];     // (2, E) int

  const int N = in_sizes[0] / F_OUT;
  const int E = in_sizes[3] / 2;
  const int* src = ei;
  const int* dst = ei + E;
  float* out = (float*)d_out;

  // Workspace layout (floats): Bmat[2048] | s_src[4N] | s_dst[4N] | rowsum[4N]
  float* Bmat   = (float*)d_ws;
  float* s_src  = Bmat + F_OUT * 16;
  float* s_dst  = s_src + (size_t)N_HEAD * N;
  float* rowsum = s_dst + (size_t)N_HEAD * N;

  hipMemsetAsync(d_out, 0, (size_t)out_size * sizeof(float), stream);
  hipMemsetAsync(rowsum, 0, (size_t)N_HEAD * N * sizeof(float), stream);

  build_b_kernel<<<1, 256, 0, stream>>>(w, a, Bmat);

  const int numTiles = (N + 15) / 16;
  const int tilesPerBlock = 8;  // 256 threads = 8 wave32
  const int gridB = (numTiles + tilesPerBlock - 1) / tilesPerBlock;
  score_wmma_kernel<<<gridB, 256, 0, stream>>>(x, Bmat, s_src, s_dst, N);

  edge_kernel<<<2048, 256, 0, stream>>>(x, w, s_src, s_dst, src, dst, rowsum,
                                        out, E, N);

  const size_t total = (size_t)N_HEAD * N * F_OUT;
  const int gridN = (int)((total + 255) / 256);
  norm_kernel<<<gridN, 256, 0, stream>>>(out, rowsum, N);
}